// listLoss_57612691309202
// MI455X (gfx1250) — compile-verified
//
#include <hip/hip_runtime.h>
#include <math.h>

// ---------------------------------------------------------------------------
// listLoss for MI455X (gfx1250, wave32).
//
// Math:  RT_target^-1 @ RT_pred = R_t^T R_p  (translations cancel), so
//        Minv = Q = R_p^T R_t is a pure rotation and
//        err^2(n,p) = v_n^T (2I - Q_p - Q_p^T) v_n = phi(v_n) . c_p,
//        phi = (x^2, y^2, z^2, xy, xz, yz), c_p = 6 coefficients.
//
// Hot pass streams the 48 MB of xyz exactly once (HBM floor ~2.1 us at
// 23.3 TB/s).  Per wave-iteration, two chained V_WMMA_F32_16X16X4_F32 compute
// err^2 for 16 points x 6 poses:  A = coefficient matrix (rows = poses,
// loaded once per wave), B = per-point features (cols = points).  D[r][n]
// lands so lanes 0-15 hold all 96 useful values in VGPRs 0-5; the tail is
// 6 bare v_sqrt_f32 (|x| via source modifier) + 6 adds per 16 points.
// ---------------------------------------------------------------------------

typedef float v2f __attribute__((ext_vector_type(2)));
typedef float v8f __attribute__((ext_vector_type(8)));

constexpr int Bn = 16;
constexpr int Nn = 262144;
constexpr int Pn = 6;

constexpr int NBLK            = 2048;
constexpr int WAVES_PER_BLOCK = 8;                         // 256 threads, wave32
constexpr int TOTAL_WAVES     = NBLK * WAVES_PER_BLOCK;    // 16384
constexpr int GROUPS_PER_B    = Nn / 16;                   // 16384 groups of 16 pts
constexpr int TOTAL_GROUPS    = Bn * GROUPS_PER_B;         // 262144
constexpr int GROUPS_PER_WAVE = TOTAL_GROUPS / TOTAL_WAVES;// 16
static_assert(GROUPS_PER_B % GROUPS_PER_WAVE == 0, "wave stays inside one batch");

// ws layout (floats)
constexpr int COEFA_OFF = 0;              // [B][4][32] A-operand VGPR images
constexpr int ROT_OFF   = Bn * 128;       // [P]        rot_per_pred (already mean)
constexpr int PART_OFF  = ROT_OFF + 8;    // [NBLK][P]  per-block pcl partials

__device__ inline void quat2R(float w, float x, float y, float z, float R[3][3]) {
  float n = sqrtf(w * w + x * x + y * y + z * z);
  w /= n; x /= n; y /= n; z /= n;
  R[0][0] = 1.f - 2.f*y*y - 2.f*z*z; R[0][1] = 2.f*x*y - 2.f*z*w;       R[0][2] = 2.f*x*z + 2.f*y*w;
  R[1][0] = 2.f*x*y + 2.f*z*w;       R[1][1] = 1.f - 2.f*x*x - 2.f*z*z; R[1][2] = 2.f*y*z - 2.f*x*w;
  R[2][0] = 2.f*x*z - 2.f*y*w;       R[2][1] = 2.f*y*z + 2.f*x*w;       R[2][2] = 1.f - 2.f*x*x - 2.f*y*y;
}

// One block, 256 threads: quadratic-form coefficients (pre-swizzled into the
// WMMA A-operand lane layout) + exact quat geodesic means.
__global__ void setup_kernel(const float* __restrict__ target_rot,  // [B,4]
                             const float* __restrict__ rot_list,    // [P,B,4]
                             float* __restrict__ coefA,             // [B][4][32]
                             float* __restrict__ rotmean) {         // [P]
  __shared__ float c_s[Bn][Pn][6];   // raw coefficients
  __shared__ float qd_s[Pn * Bn];
  const int tid = threadIdx.x;

  if (tid < Pn * Bn) {
    const int b = tid & 15;
    const int p = tid >> 4;

    const float tw = target_rot[b * 4 + 0], tx = target_rot[b * 4 + 1];
    const float ty = target_rot[b * 4 + 2], tz = target_rot[b * 4 + 3];
    const int   qi = (p * Bn + b) * 4;
    const float pw = rot_list[qi + 0], qx = rot_list[qi + 1];
    const float qy = rot_list[qi + 2], qz = rot_list[qi + 3];

    float Rt[3][3], Rp[3][3];
    quat2R(tw, tx, ty, tz, Rt);
    quat2R(pw, qx, qy, qz, Rp);

    float Q[3][3];                                  // Q = Rp^T * Rt
    for (int i = 0; i < 3; ++i)
      for (int j = 0; j < 3; ++j)
        Q[i][j] = Rp[0][i]*Rt[0][j] + Rp[1][i]*Rt[1][j] + Rp[2][i]*Rt[2][j];

    c_s[b][p][0] = 2.0f - 2.0f * Q[0][0];           // x^2
    c_s[b][p][1] = 2.0f - 2.0f * Q[1][1];           // y^2
    c_s[b][p][2] = 2.0f - 2.0f * Q[2][2];           // z^2
    c_s[b][p][3] = -2.0f * (Q[0][1] + Q[1][0]);     // xy
    c_s[b][p][4] = -2.0f * (Q[0][2] + Q[2][0]);     // xz
    c_s[b][p][5] = -2.0f * (Q[1][2] + Q[2][1]);     // yz

    // quat_distance(rot_list, target_rot): t = q * conj(r)
    const float t0 = tw*pw + tx*qx + ty*qy + tz*qz;
    const float t1 = tw*qx - tx*pw + ty*qz - tz*qy;
    const float t2 = tw*qy - tx*qz - ty*pw + tz*qx;
    const float t3 = tw*qz + tx*qy - ty*qx - tz*pw;
    qd_s[p * Bn + b] = 2.0f * atan2f(sqrtf(t1*t1 + t2*t2 + t3*t3), fabsf(t0));
  }
  __syncthreads();

  // Emit A-operand per-lane images: q=0..3 -> {a1.x, a1.y, a2.x, a2.y}.
  // A layout (16x4 f32): lane m -> K0,K1 ; lane m+16 -> K2,K3.
  // wmma1 uses coeffs k = 0,1 | 2,3 ; wmma2 uses k = 4,5 | zero,zero.
  for (int i = tid; i < Bn * 128; i += blockDim.x) {
    const int b    = i >> 7;
    const int q    = (i >> 5) & 3;
    const int lane = i & 31;
    const int m    = lane & 15;          // pose (A row)
    const bool hi  = lane >= 16;
    float v = 0.0f;
    if (m < Pn) {
      if      (q == 0) v = c_s[b][m][hi ? 2 : 0];
      else if (q == 1) v = c_s[b][m][hi ? 3 : 1];
      else if (q == 2) v = hi ? 0.0f : c_s[b][m][4];
      else             v = hi ? 0.0f : c_s[b][m][5];
    }
    coefA[i] = v;
  }

  if (tid < Pn) {                                   // deterministic mean over b
    float s = 0.0f;
    for (int b = 0; b < Bn; ++b) s += qd_s[tid * Bn + b];
    rotmean[tid] = s / (float)Bn;
  }
}

// Streaming pass: per wave, 16 points/iter; two WMMAs -> err^2 for 16 pts x 6 poses.
__global__ __launch_bounds__(256) void pcl_err_kernel(const float* __restrict__ pc,    // [B,4,N]
                                                      const float* __restrict__ coefA, // [B][4][32]
                                                      float* __restrict__ part) {      // [NBLK][P]
  __shared__ float wacc[WAVES_PER_BLOCK * Pn];
  const int  tid  = threadIdx.x;
  const int  lane = tid & 31;
  const int  wave = tid >> 5;
  const int  col  = lane & 15;          // point-in-group (B-matrix column)
  const bool hi   = lane >= 16;         // half-wave holds K=2,3

  const int gwave = blockIdx.x * WAVES_PER_BLOCK + wave;
  const int g0    = gwave * GROUPS_PER_WAVE;
  const int b     = g0 / GROUPS_PER_B;  // constant for this wave

  // A operand (pose coefficients), loaded once per wave.
  const float* ca = coefA + b * 128;
  v2f am1, am2;
  am1.x = ca[0 * 32 + lane];
  am1.y = ca[1 * 32 + lane];
  am2.x = ca[2 * 32 + lane];
  am2.y = ca[3 * 32 + lane];

  const float* px = pc + (size_t)b * 4u * (size_t)Nn;
  const float* py = px + Nn;
  const float* pz = px + 2 * Nn;
  const int n0 = (g0 % GROUPS_PER_B) * 16 + col;

  float acc[Pn] = {0.f, 0.f, 0.f, 0.f, 0.f, 0.f};
  for (int k = 0; k < GROUPS_PER_WAVE; ++k) {
    const int n = n0 + k * 16;
    const float x = __builtin_nontemporal_load(px + n);   // streamed once: TH=NT
    const float y = __builtin_nontemporal_load(py + n);
    const float z = __builtin_nontemporal_load(pz + n);

    // B operand (4x16 f32): lane n -> rows 0,1 ; lane n+16 -> rows 2,3.
    // wmma1 features (x^2, y^2 | z^2, xy); wmma2 features (xz, yz | 0, 0).
    v2f bm1, bm2;
    bm1.x = (hi ? z : x) * (hi ? z : x);
    bm1.y = (hi ? x : y) * y;
    bm2.x = (hi ? 0.0f : x) * z;
    bm2.y = (hi ? 0.0f : y) * z;

    v8f c = {};
    c = __builtin_amdgcn_wmma_f32_16x16x4_f32(false, am1, false, bm1, (short)0, c, false, false);
    c = __builtin_amdgcn_wmma_f32_16x16x4_f32(false, am2, false, bm2, (short)0, c, false, false);

    // D[r][n]: lane n (n<16) holds err^2 for pose r in VGPR r (r<6).
    // Bare v_sqrt_f32 with |x| source modifier: no IEEE slow path, no clamp.
#pragma unroll
    for (int r = 0; r < Pn; ++r)
      acc[r] += __builtin_amdgcn_sqrtf(__builtin_fabsf(c[r]));
  }

  // Wave reduction over the 16 point lanes (lanes 16-31 hold exact zeros
  // from the zero A rows, but are excluded anyway).
#pragma unroll
  for (int r = 0; r < Pn; ++r) {
    float a = acc[r];
    a += __shfl_down(a, 8, 32);
    a += __shfl_down(a, 4, 32);
    a += __shfl_down(a, 2, 32);
    a += __shfl_down(a, 1, 32);
    acc[r] = a;
  }
  if (lane == 0) {
#pragma unroll
    for (int r = 0; r < Pn; ++r) wacc[wave * Pn + r] = acc[r];
  }
  __syncthreads();

  if (tid < Pn) {                                  // deterministic per-block sum
    float s = 0.0f;
    for (int w = 0; w < WAVES_PER_BLOCK; ++w) s += wacc[w * Pn + tid];
    part[blockIdx.x * Pn + tid] = s;
  }
}

__global__ void finalize_kernel(const float* __restrict__ part,     // [NBLK][P]
                                const float* __restrict__ rotmean,  // [P]
                                float* __restrict__ out) {          // [3]
  __shared__ float pcl_s[Pn];
  const int tid = threadIdx.x;
  if (tid < Pn) {
    float s = 0.0f;
    for (int i = 0; i < NBLK; ++i) s += part[i * Pn + tid];
    pcl_s[tid] = s / (float)Nn;                    // mean over N, sum over B
  }
  __syncthreads();
  if (tid == 0) {
    float total = 0.0f, rl = 0.0f, pl = 0.0f;
    for (int p = 0; p < Pn; ++p) {
      const float w    = powf(0.8f, (float)(Pn - 1 - p));  // GAMMA^(P-1-p)
      const float rot  = rotmean[p];                       // RESCALE_ROT = 1
      const float pclB = pcl_s[p] / (float)Bn;
      total += w * (0.5f * rot + 0.5f * pclB);             // WEIGHT_PC = 0.5
      rl    += w * rot;
      pl    += w * pclB;
    }
    out[0] = total; out[1] = rl; out[2] = pl;
  }
}

extern "C" void kernel_launch(void* const* d_in, const int* in_sizes, int n_in,
                              void* d_out, int out_size, void* d_ws, size_t ws_size,
                              hipStream_t stream) {
  (void)in_sizes; (void)n_in; (void)out_size; (void)ws_size;
  const float* pc = (const float*)d_in[0];   // [B,4,N]
  // d_in[1] = target_transl: cancels analytically (Minv is a pure rotation).
  const float* tr = (const float*)d_in[2];   // [B,4]
  const float* rl = (const float*)d_in[3];   // [P,B,4]

  float* ws      = (float*)d_ws;
  float* coefA   = ws + COEFA_OFF;
  float* rotmean = ws + ROT_OFF;
  float* part    = ws + PART_OFF;

  setup_kernel<<<1, 256, 0, stream>>>(tr, rl, coefA, rotmean);
  pcl_err_kernel<<<NBLK, 256, 0, stream>>>(pc, coefA, part);
  finalize_kernel<<<1, 32, 0, stream>>>(part, rotmean, (float*)d_out);
}